// TriplesDistances_841813590532
// MI455X (gfx1250) — compile-verified
//
#include <hip/hip_runtime.h>

// Problem constants (match the reference)
#define BB 8
#define AA 256
#define TT 496
#define NN 32

typedef __attribute__((ext_vector_type(2))) float v2f;
typedef __attribute__((ext_vector_type(8))) float v8f;

// One workgroup per (b,a). 256 threads = 8 wave32 waves.
__global__ __launch_bounds__(256) void TriplesDistances_kernel(
    const float* __restrict__ positions,    // [B,A,3]
    const int*   __restrict__ nbr_j,        // [B,A,T]
    const int*   __restrict__ nbr_k,        // [B,A,T]
    const int*   __restrict__ offidx_j,     // [B,A,T]
    const int*   __restrict__ offidx_k,     // [B,A,T]
    const float* __restrict__ cell,         // [B,3,3]
    const float* __restrict__ cell_offsets, // [B,A,N,3]
    float*       __restrict__ out)          // [3][B,A,T] concatenated
{
    // positions of batch b, padded to stride 4 for clean LDS addressing
    __shared__ float s_pos[AA * 4];
    // per-(b,a) PBC offset table: offsets[n][e] = sum_k co[n][k]*cell[k][e]
    __shared__ float s_off[NN * 4];

    const int ba  = blockIdx.x;     // b*A + a
    const int b   = ba / AA;
    const int a   = ba - b * AA;
    const int tid = threadIdx.x;

    // ---- Stage positions[b] into LDS (768 floats, coalesced) ----
    const float* posb = positions + (size_t)b * AA * 3;
    for (int i = tid; i < AA * 3; i += 256) {
        s_pos[(i / 3) * 4 + (i % 3)] = posb[i];
    }

    // ---- Wave 0: offsets = cell_offsets[b,a] (32x3) @ cell[b] (3x3)  ----
    // Two V_WMMA_F32_16X16X4_F32 ops: M tiles of 16, K padded 3->4, N padded 3->16.
    if (tid < 32) {
        const int lane = tid;
        const int half = lane >> 4;   // 0: lanes 0-15, 1: lanes 16-31
        const int col  = lane & 15;   // N column for B/C/D, row M for A

        // B matrix 4x16 (f32, 2 VGPRs): lanes 0-15 hold K=0 (v0), K=1 (v1);
        // lanes 16-31 hold K=2 (v0), K=3 (v1, zero pad). N = lane&15, cols >=3 zero.
        v2f bm;
        float b0 = 0.0f, b1 = 0.0f;
        if (col < 3) {
            b0 = cell[b * 9 + (2 * half) * 3 + col];       // K = 0 or 2
            if (half == 0) b1 = cell[b * 9 + 1 * 3 + col]; // K = 1 (K=3 pads to 0)
        }
        bm[0] = b0; bm[1] = b1;

        const float* co_base = cell_offsets + ((size_t)ba * NN) * 3;

        #pragma unroll
        for (int m_base = 0; m_base < NN; m_base += 16) {
            // A matrix 16x4 (f32, 2 VGPRs): row M = lane&15 for both halves;
            // lanes 0-15: K=0 (v0), K=1 (v1); lanes 16-31: K=2 (v0), K=3 (v1 pad).
            const float* co = co_base + (size_t)(m_base + col) * 3;
            v2f am;
            am[0] = co[2 * half];                       // K = 0 or 2 (both < 3)
            am[1] = (half == 0) ? co[1] : 0.0f;         // K = 1, or K=3 -> 0
            v8f c = {};
            // 8 args: (neg_a, A, neg_b, B, c_mod, C, reuse_a, reuse_b)
            v8f d = __builtin_amdgcn_wmma_f32_16x16x4_f32(
                false, am, false, bm, (short)0, c, false, false);
            // D layout: lane group g=half -> rows M = r + 8*g, N = lane&15.
            if (col < 3) {
                #pragma unroll
                for (int r = 0; r < 8; ++r) {
                    s_off[(m_base + 8 * half + r) * 4 + col] = d[r];
                }
            }
        }
    }
    __syncthreads();

    // ---- Per-triple distance computation (memory-bound streaming part) ----
    const float pix = s_pos[a * 4 + 0];
    const float piy = s_pos[a * 4 + 1];
    const float piz = s_pos[a * 4 + 2];

    const size_t base  = (size_t)ba * TT;
    const size_t plane = (size_t)BB * AA * TT;

    for (int t = tid; t < TT; t += 256) {
        const size_t idx = base + t;
        // One-touch streaming reads: non-temporal so they don't pollute WGP$
        const int j  = __builtin_nontemporal_load(&nbr_j[idx]);
        const int k  = __builtin_nontemporal_load(&nbr_k[idx]);
        const int oj = __builtin_nontemporal_load(&offidx_j[idx]);
        const int ok = __builtin_nontemporal_load(&offidx_k[idx]);

        const float pjx = s_pos[j * 4 + 0] + s_off[oj * 4 + 0];
        const float pjy = s_pos[j * 4 + 1] + s_off[oj * 4 + 1];
        const float pjz = s_pos[j * 4 + 2] + s_off[oj * 4 + 2];
        const float pkx = s_pos[k * 4 + 0] + s_off[ok * 4 + 0];
        const float pky = s_pos[k * 4 + 1] + s_off[ok * 4 + 1];
        const float pkz = s_pos[k * 4 + 2] + s_off[ok * 4 + 2];

        const float dijx = pjx - pix, dijy = pjy - piy, dijz = pjz - piz;
        const float dikx = pkx - pix, diky = pky - piy, dikz = pkz - piz;
        const float djkx = pjx - pkx, djky = pjy - pky, djkz = pjz - pkz;

        const float rij = sqrtf(dijx * dijx + dijy * dijy + dijz * dijz) + 1e-9f;
        const float rik = sqrtf(dikx * dikx + diky * diky + dikz * dikz) + 1e-9f;
        const float rjk = sqrtf(djkx * djkx + djky * djky + djkz * djkz) + 1e-9f;

        // One-touch streaming writes (12 MB total): non-temporal
        __builtin_nontemporal_store(rij, &out[idx]);
        __builtin_nontemporal_store(rik, &out[plane + idx]);
        __builtin_nontemporal_store(rjk, &out[2 * plane + idx]);
    }
}

extern "C" void kernel_launch(void* const* d_in, const int* in_sizes, int n_in,
                              void* d_out, int out_size, void* d_ws, size_t ws_size,
                              hipStream_t stream) {
    const float* positions    = (const float*)d_in[0];
    const int*   nbr_j        = (const int*)d_in[1];
    const int*   nbr_k        = (const int*)d_in[2];
    const int*   offidx_j     = (const int*)d_in[3];
    const int*   offidx_k     = (const int*)d_in[4];
    const float* cell         = (const float*)d_in[5];
    const float* cell_offsets = (const float*)d_in[6];
    float*       out          = (float*)d_out;

    dim3 grid(BB * AA);   // 2048 workgroups, one per (b,a)
    dim3 block(256);      // 8 wave32 waves
    TriplesDistances_kernel<<<grid, block, 0, stream>>>(
        positions, nbr_j, nbr_k, offidx_j, offidx_k, cell, cell_offsets, out);
}